// MultiLevelROIVisualPrompt_17051020165121
// MI455X (gfx1250) — compile-verified
//
#include <hip/hip_runtime.h>
#include <hip/hip_bf16.h>

typedef __attribute__((ext_vector_type(2))) float v2f;
typedef __attribute__((ext_vector_type(8))) float v8f;

#define OUT_SIZE   7
#define SR         2
#define SPP        (OUT_SIZE * SR)            // 14 samples per dim
#define NSAMP      (SPP * SPP)                // 196 sample points per box
#define NCHUNK     (NSAMP / 4)                // 49 WMMA K=4 chunks
#define KBOX       64
#define CTOT       2880
#define CATHW      192
#define SCALE      0.25f

// Composed (resize o ROI-bilinear) weights along one axis, collapsed onto at
// most 3 consecutive native indices starting at n0. The two ROI corners c0,
// c1 = c0+1 on the 192-grid map to native coords r apart (r <= 0.5), so their
// tent supports overlap; the union spans <= 3 native taps. Branchless slot
// accumulation via cndmask selects.
template <int N>
__device__ __forceinline__ void axis_weights(int c0, int c1, float l,
                                             int& n0, float* w) {
    constexpr float r   = (float)N * (1.0f / 192.0f);
    constexpr float off = 0.5f * r - 0.5f;
    constexpr float nm1 = (float)(N - 1);
    float v0 = fminf(fmaxf((float)c0 * r + off, 0.0f), nm1);
    float v1 = fminf(fmaxf((float)c1 * r + off, 0.0f), nm1);
    int a0 = (int)v0; float f0 = v0 - (float)a0;       // corner-0 native base
    int b0 = (int)v1; float f1 = v1 - (float)b0;       // corner-1 native base
    int a1 = a0 + 1 < N ? a0 + 1 : N - 1;
    int b1 = b0 + 1 < N ? b0 + 1 : N - 1;
    float h = 1.0f - l;
    float wa1 = h * f0;                                // tap a1
    float wb0 = l * (1.0f - f1);                       // tap b0
    float wb1 = l * f1;                                // tap b1
    int ia1 = a1 - a0;                                 // 0 or 1
    int ib0 = b0 - a0;                                 // 0 or 1 (v1-v0 = r < 1)
    int ib1 = b1 - a0;                                 // 0, 1 or 2
    n0 = a0;
    w[0] = h * (1.0f - f0)
         + (ia1 == 0 ? wa1 : 0.0f)
         + (ib0 == 0 ? wb0 : 0.0f)
         + (ib1 == 0 ? wb1 : 0.0f);
    w[1] = (ia1 == 1 ? wa1 : 0.0f)
         + (ib0 == 1 ? wb0 : 0.0f)
         + (ib1 == 1 ? wb1 : 0.0f);
    w[2] = (ib1 == 2 ? wb1 : 0.0f);
}

// ROI-align bilinear sample of the virtual 192x192 (possibly resized) channel
// plane at clamped coords (yc, xc). For N != 192 the bilinear upsample
// (half-pixel centers, edge clamp) is composed with the ROI bilinear as a
// separable 3x3 native-tap stencil: 9 gathers instead of 16, and the 425 MB
// concat map is never materialized — all gathers stay in the 53 MB of
// L2-resident native maps.
template <int N>
__device__ __forceinline__ float sample_cat(const float* __restrict__ fc,
                                            float yc, float xc) {
    int y0 = (int)yc, x0 = (int)xc;
    int y1 = y0 + 1 < CATHW ? y0 + 1 : CATHW - 1;
    int x1 = x0 + 1 < CATHW ? x0 + 1 : CATHW - 1;
    float ly = yc - (float)y0, lx = xc - (float)x0;
    if constexpr (N == CATHW) {
        float v00 = fc[y0 * CATHW + x0], v01 = fc[y0 * CATHW + x1];
        float v10 = fc[y1 * CATHW + x0], v11 = fc[y1 * CATHW + x1];
        float top = v00 + lx * (v01 - v00);
        float bot = v10 + lx * (v11 - v10);
        return top + ly * (bot - top);
    } else {
        int ny0, nx0;
        float wy[3], wx[3];
        axis_weights<N>(y0, y1, ly, ny0, wy);
        axis_weights<N>(x0, x1, lx, nx0, wx);
        int cx1 = nx0 + 1 < N ? nx0 + 1 : N - 1;
        int cx2 = nx0 + 2 < N ? nx0 + 2 : N - 1;
        float s = 0.0f;
#pragma unroll
        for (int iy = 0; iy < 3; ++iy) {
            int ry = ny0 + iy < N ? ny0 + iy : N - 1;
            const float* row = fc + ry * N;
            float rv = wx[0] * row[nx0] + wx[1] * row[cx1] + wx[2] * row[cx2];
            s += wy[iy] * rv;
        }
        return s;
    }
}

// Branchless sample: always gather (clamped coords are safe), then cndmask
// with the validity predicate — no exec save/restore per sample point.
template <int N>
__device__ __forceinline__ float sample_value(const float* __restrict__ fc,
                                              float y1v, float x1v,
                                              float bin_h, float bin_w, int s) {
    int sy = s / SPP;
    int sx = s - sy * SPP;
    float yy = y1v + (0.25f + 0.5f * (float)sy) * bin_h;
    float xx = x1v + (0.25f + 0.5f * (float)sx) * bin_w;
    bool valid = (yy > -1.0f) & (yy < 192.0f) & (xx > -1.0f) & (xx < 192.0f);
    float yc = fminf(fmaxf(yy, 0.0f), 191.0f);
    float xc = fminf(fmaxf(xx, 0.0f), 191.0f);
    float v = sample_cat<N>(fc, yc, xc);
    return valid ? v : 0.0f;
}

// One wave (32 lanes) per (box, 16-channel tile), specialized per pyramid
// level. The 196-sample mean runs on the matrix engine: per K=4 chunk,
// A = V[16 ch x 4 samples] (lanes 0-15 carry K=0,1; lanes 16-31 carry K=2,3),
// B = 1/196 splat; 49 accumulating V_WMMA_F32_16X16X4_F32 steps leave the 16
// channel means in column N=0 of D (lanes 0 and 16, VGPRs 0..7).
template <int N>
__global__ __launch_bounds__(256) void roi_wmma_level(
    const float* __restrict__ feat, const float* __restrict__ boxes,
    float* __restrict__ out, int c0base, int ntile) {
    int wave = threadIdx.x >> 5;
    int lane = threadIdx.x & 31;
    int job  = blockIdx.x * 8 + wave;
    if (job >= KBOX * ntile) return;

    int k  = job / ntile;
    int t  = job - k * ntile;

    int cloc = lane & 15;                              // channel within tile
    const float* fc = feat + (size_t)(t * 16 + cloc) * (size_t)(N * N);
    __builtin_prefetch(fc, 0, 1);                      // global_prefetch_b8

    // Box geometry (same 4 loads in every lane -> broadcast).
    float bx1 = boxes[k * 4 + 0] * SCALE;
    float by1 = boxes[k * 4 + 1] * SCALE;
    float bx2 = boxes[k * 4 + 2] * SCALE;
    float by2 = boxes[k * 4 + 3] * SCALE;
    float roi_w = fmaxf(bx2 - bx1, 1.0f);
    float roi_h = fmaxf(by2 - by1, 1.0f);
    float bin_w = roi_w * (1.0f / OUT_SIZE);
    float bin_h = roi_h * (1.0f / OUT_SIZE);

    int hi = lane >> 4;                                // A-matrix K half

    v2f bsum; bsum.x = 1.0f / (float)NSAMP; bsum.y = 1.0f / (float)NSAMP;
    v8f acc = {};

    for (int ci = 0; ci < NCHUNK; ++ci) {
        int sA = ci * 4 + hi * 2;
        v2f a;
        a.x = sample_value<N>(fc, by1, bx1, bin_h, bin_w, sA);
        a.y = sample_value<N>(fc, by1, bx1, bin_h, bin_w, sA + 1);
        acc = __builtin_amdgcn_wmma_f32_16x16x4_f32(
            /*neg_a=*/false, a, /*neg_b=*/false, bsum,
            /*c_mod=*/(short)0, acc, /*reuse_a=*/false, /*reuse_b=*/false);
    }

    float* o = out + (size_t)k * CTOT + (size_t)(c0base + t * 16);
    if (lane == 0) {
#pragma unroll
        for (int r = 0; r < 8; ++r) o[r] = acc[r];
    } else if (lane == 16) {
#pragma unroll
        for (int r = 0; r < 8; ++r) o[8 + r] = acc[r];
    }
}

// out[k, c] += sine positional embedding (order: cy, cx, nw, nh; 720 dims each).
__global__ __launch_bounds__(256) void posembed_kernel(
    const float* __restrict__ boxes, float* __restrict__ out) {
    int idx = blockIdx.x * blockDim.x + threadIdx.x;
    if (idx >= KBOX * CTOT) return;
    int k = idx / CTOT;
    int c = idx - k * CTOT;
    int q = c / 720;
    int j = c - q * 720;

    const float inv_img = 1.0f / 768.0f;               // W / SPATIAL_SCALE
    float x1 = boxes[k * 4 + 0] * inv_img;
    float y1 = boxes[k * 4 + 1] * inv_img;
    float nw = boxes[k * 4 + 2] * inv_img - x1;
    float nh = boxes[k * 4 + 3] * inv_img - y1;
    float cx = x1 + 0.5f * nw;
    float cy = y1 + 0.5f * nh;
    float v = (q == 0) ? cy : (q == 1) ? cx : (q == 2) ? nw : nh;

    float e = (float)(2 * (j >> 1)) * (1.0f / 720.0f); // dim_t exponent
    float dimt = __powf(10000.0f, e);
    float p = v * 6.28318530717958647692f / dimt;
    float s = (j & 1) ? __cosf(p) : __sinf(p);
    out[idx] += s;
}

extern "C" void kernel_launch(void* const* d_in, const int* in_sizes, int n_in,
                              void* d_out, int out_size, void* d_ws, size_t ws_size,
                              hipStream_t stream) {
    (void)in_sizes; (void)n_in; (void)d_ws; (void)ws_size; (void)out_size;
    const float* feat0 = (const float*)d_in[0];
    const float* feat1 = (const float*)d_in[1];
    const float* feat2 = (const float*)d_in[2];
    const float* feat3 = (const float*)d_in[3];
    const float* boxes = (const float*)d_in[4];
    float* out = (float*)d_out;

    // Per-level launches: (channels/16) tiles x 64 boxes, 8 waves per block.
    // 12/24/48/96 tiles -> 768/1536/3072/6144 wave-jobs, all multiples of 8.
    roi_wmma_level<192><<<(KBOX * 12) / 8, 256, 0, stream>>>(feat0, boxes, out,    0, 12);
    roi_wmma_level< 96><<<(KBOX * 24) / 8, 256, 0, stream>>>(feat1, boxes, out,  192, 24);
    roi_wmma_level< 48><<<(KBOX * 48) / 8, 256, 0, stream>>>(feat2, boxes, out,  576, 48);
    roi_wmma_level< 24><<<(KBOX * 96) / 8, 256, 0, stream>>>(feat3, boxes, out, 1344, 96);

    int n = KBOX * CTOT;
    posembed_kernel<<<(n + 255) / 256, 256, 0, stream>>>(boxes, out);
}